// CombNewsvendorLayer_61400852464229
// MI455X (gfx1250) — compile-verified
//
#include <hip/hip_runtime.h>

// CombNewsvendorLayer for MI455X (gfx1250).
//
// Bandwidth-bound kernel (~318 MB total traffic => ~14us at 23.3 TB/s).
// The 60-iter bisection is replaced by an exact closed form:
//   g(z) = (cdf(z) - CF*P) - GAMMA*(M1 - z*M0)/sqrt(M2 - 2 z M1 + z^2 M0)
// is piecewise-smooth (cdf constant between support points); on the
// bracketing interval g(z)=0 reduces to a quadratic whose valid root is
//   z = (M1*D + a*sqrt(D*(M1^2 - M0*M2))) / (M0*D),  D = a^2 - g^2*M0, a = c - CF*P
// (discriminant >= 0 by Cauchy-Schwarz; single branch matches sign(s)=sign(a)).
//
// CDNA5-specific path: expert tiles are staged with GLOBAL_LOAD_ASYNC_TO_LDS_B32
// (ASYNCcnt-tracked DMA, no VGPR landing), scattered into a transposed
// [col][row] LDS layout with stride 129 dwords -> conflict-free for both the
// DMA writes and the per-thread column scans. WMMA is deliberately not used:
// the only matmul-shaped op has K=3 and the kernel sits at ~4 flop/byte.

#define CFRAC   0.6f
#define GAMMA_C 0.1f
#define SSUP    101        // support size |Y|
#define PS      129        // LDS row stride in dwords (odd -> conflict-free)
#define ROWS    128        // rows per block == blockDim.x (4 waves, wave32)
#define NEXP    3

__device__ __forceinline__ void async_ld_b32(unsigned lds_byte_addr,
                                             unsigned gofs_bytes,
                                             const float* sbase) {
  // LDS byte address from vdst; global address = sbase + 32-bit byte offset
  asm volatile("global_load_async_to_lds_b32 %0, %1, %2"
               :: "v"(lds_byte_addr), "v"(gofs_bytes), "s"(sbase)
               : "memory");
}

extern "C" __global__ __launch_bounds__(ROWS)
void cnv_kernel(const float* __restrict__ X,   // [3, B, S]
                const float* __restrict__ W,   // [3]
                float* __restrict__ out,       // z[B] | pin[B*S] | e[B*S] | w[B*S]
                int B)
{
  __shared__ float xT[NEXP * SSUP * PS];   // 3 transposed tiles; tile0 becomes p
  __shared__ float zrow[ROWS];

  const int tid = threadIdx.x;
  const int r0  = blockIdx.x * ROWS;
  const size_t BS = (size_t)B * SSUP;

  const unsigned l0   = (unsigned)(size_t)(&xT[0]);   // low 32 bits = LDS offset
  const unsigned TILE = (unsigned)(SSUP * PS * 4);

  // ---- Phase 1: async DMA stage of 3 expert tiles, transposed into LDS ----
  // 128*101 = 12928 elements per tile; contiguous global reads (coalesced),
  // scattered LDS writes implement the transpose.
  for (int idx = tid; idx < ROWS * SSUP; idx += ROWS) {
    int row = idx / SSUP;
    int col = idx - row * SSUP;
    int grow = r0 + row; if (grow >= B) grow = B - 1;   // tail guard (dup load)
    unsigned gofs = (unsigned)(((unsigned)grow * SSUP + (unsigned)col) * 4u);
    unsigned lofs = (unsigned)((col * PS + row) * 4);
    async_ld_b32(l0 +            lofs, gofs, X);
    async_ld_b32(l0 + TILE +     lofs, gofs, X + BS);
    async_ld_b32(l0 + 2u*TILE +  lofs, gofs, X + 2*BS);
  }
  asm volatile("s_wait_asynccnt 0" ::: "memory");
  __syncthreads();

  // ---- Phase 2: fused expert-combine + moment scan (thread r owns row r) ----
  const float w0 = W[0], w1 = W[1], w2 = W[2];
  const int r = tid;
  float P = 0.f, M0 = 0.f, M1 = 0.f, M2 = 0.f;
  float* pT = xT;                                 // p overwrites tile 0 in place
  const float* x1T = xT + SSUP * PS;
  const float* x2T = xT + 2 * SSUP * PS;
  for (int i = 0; i < SSUP; ++i) {
    int o = i * PS + r;                           // lane-consecutive -> no conflicts
    float v = fmaf(w0, xT[o], fmaf(w1, x1T[o], w2 * x2T[o]));
    pT[o] = v;
    float y  = (float)i * 0.01f;                  // Y_i (<=1 ulp vs reference table)
    float qq = v * v;
    P  += v;
    M0 += qq;
    M1  = fmaf(qq, y, M1);
    M2  = fmaf(qq, y * y, M2);
  }
  // pT column r is read only by thread r below -> no barrier needed yet.

  // ---- Phase 3: bracket scan + closed-form root (branchless) ----
  const float A  = CFRAC * P;
  const float g2 = GAMMA_C * GAMMA_C;
  int   kf = SSUP;          // first k with g(Y_k^-) >= 0 ; SSUP => root at z=1
  float cInt = 0.f;         // cdf on the bracketing interval
  int   prevW = 0;          // g(Y_{k-1}^+) >= 0  => jump crossing at grid point
  int   found = 0, prevGE = 0;
  float c = 0.f;
  for (int k = 0; k < SSUP; ++k) {
    float y  = (float)k * 0.01f;
    float pk = pT[k * PS + r];
    float s  = fmaf(-y, M0, M1);                        // M1 - y*M0
    float qz = fmaf(y, fmaf(y, M0, -2.0f * M1), M2);    // ||p(Y-y)||^2
    float n  = fmaxf(sqrtf(fmaxf(qz, 0.f)), 1e-12f);
    float a  = c - A;
    int ge_l = (a * n >= GAMMA_C * s);          // sign of g(Y_k^-): avoids the divide
    int ge_r = ((a + pk) * n >= GAMMA_C * s);   // sign of g(Y_k^+)
    int trig = ge_l & (found ^ 1);
    kf    = trig ? k      : kf;
    cInt  = trig ? c      : cInt;
    prevW = trig ? prevGE : prevW;
    found |= ge_l;
    prevGE = ge_r;
    c += pk;
  }
  float z;
  if (kf >= SSUP) {
    z = 1.0f;                                    // g < 0 everywhere
  } else if (kf == 0) {
    z = 0.0f;                                    // g >= 0 already at z=0
  } else if (prevW) {
    z = (float)(kf - 1) * 0.01f;                 // g jumps across 0 at Y_{k-1}
  } else {
    float zl = (float)(kf - 1) * 0.01f, zr = (float)kf * 0.01f;
    float a    = cInt - A;
    float D    = fmaf(a, a, -g2 * M0);                        // <= 0 near root
    float disc = fmaxf(D * fmaf(M1, M1, -M0 * M2), 0.f);      // >= 0 (Cauchy-Schwarz)
    float den  = D * M0;
    float zq   = (fabsf(den) > 1e-30f)
               ? (fmaf(M1, D, a * sqrtf(disc)) / den)
               : 0.5f * (zl + zr);
    z = fminf(fmaxf(zq, zl), zr);
  }
  zrow[r] = z;
  if (r0 + r < B) out[r0 + r] = z;               // z[:,None] output, coalesced
  __syncthreads();                                // publish zrow + full pT tile

  // ---- Phase 4: cooperative coalesced streaming of pin / e / w_err ----
  float* outPin = out + B;
  float* outE   = outPin + BS;
  float* outW   = outE + BS;
  for (int idx = tid; idx < ROWS * SSUP; idx += ROWS) {
    int row = idx / SSUP;
    int col = idx - row * SSUP;
    if (r0 + row < B) {
      float zz  = zrow[row];
      float pv  = pT[col * PS + row];
      float e   = (float)col * 0.01f - zz;
      float pin = fmaxf(CFRAC * e, (CFRAC - 1.0f) * e);
      size_t o  = (size_t)r0 * SSUP + (size_t)idx;   // == (r0+row)*S + col
      outPin[o] = pin;
      outE[o]   = e;
      outW[o]   = pv * e;
    }
  }
}

extern "C" void kernel_launch(void* const* d_in, const int* in_sizes, int n_in,
                              void* d_out, int out_size, void* d_ws, size_t ws_size,
                              hipStream_t stream) {
  (void)n_in; (void)out_size; (void)d_ws; (void)ws_size;
  const float* X = (const float*)d_in[0];   // [3, B, 101]
  const float* W = (const float*)d_in[1];   // [3]
  float* out = (float*)d_out;
  int B = in_sizes[0] / (NEXP * SSUP);
  int grid = (B + ROWS - 1) / ROWS;
  cnv_kernel<<<dim3(grid), dim3(ROWS), 0, stream>>>(X, W, out, B);
}